// ElboFB_47364899340954
// MI455X (gfx1250) — compile-verified
//
#include <hip/hip_runtime.h>
#include <math.h>

// ---------------- problem constants (fixed by setup_inputs) ----------------
constexpr int W  = 512;
constexpr int H  = 384;
constexpr int NB = 8;                 // batch
constexpr long HW = (long)H * W;

// tile geometry for the energy kernel
constexpr int TW = 32, TH = 16, HL = 2;
constexpr int SW = TW + 2 * HL;       // 36
constexpr int SH = TH + 2 * HL;       // 20

// 5-tap gradient filter taps (k[2] == 0, skipped)
constexpr float K0 = -1.0f / 12.0f;
constexpr float K1 =  2.0f / 3.0f;
constexpr float K3 = -2.0f / 3.0f;
constexpr float K4 =  1.0f / 12.0f;

// grid sizes
constexpr int TILES_X = W / TW;       // 16
constexpr int TILES_Y = H / TH;       // 24
constexpr int NTILES  = TILES_X * TILES_Y * NB;   // 3072 per direction
constexpr int NPART_E = 2 * NTILES;               // 6144 (both directions)
constexpr int NPART_M = NTILES;                   // 3072 (epe / entropy)

typedef float v2f __attribute__((ext_vector_type(2)));
typedef float v8f __attribute__((ext_vector_type(8)));

__device__ __forceinline__ float sigf(float x) { return 1.0f / (1.0f + __expf(-x)); }
__device__ __forceinline__ float penf(float x) { return sqrtf(x + 1e-5f); }

// async global->LDS copy of one dword (CDNA5 GLOBAL_LOAD_ASYNC_TO_LDS_B32,
// tracked by ASYNCcnt). lds_off = LDS byte offset (low 32 bits of the
// shared-aperture flat address); gaddr = 64-bit global VA.
__device__ __forceinline__ void async_copy_b32(unsigned lds_off, const float* gp) {
  unsigned long long ga = (unsigned long long)(size_t)gp;
  asm volatile("global_load_async_to_lds_b32 %0, %1, off"
               :: "v"(lds_off), "v"(ga) : "memory");
}
__device__ __forceinline__ void async_wait0() {
  asm volatile("s_wait_asynccnt 0" ::: "memory");
}
__device__ __forceinline__ unsigned lds_off_of(const void* p) {
  return (unsigned)(size_t)p;   // low 32 bits of LDS-aperture flat address == LDS byte offset
}

// bilinear tap positions/weights, zero-weight for out-of-image taps (matches reference)
__device__ __forceinline__ void bilin_taps(float xq, float yq, int* xs, int* ys, float* ws) {
  float x0 = floorf(xq), y0 = floorf(yq);
  float wx1 = xq - x0, wy1 = yq - y0;
  float wx0 = 1.0f - wx1, wy0 = 1.0f - wy1;
#pragma unroll
  for (int t = 0; t < 4; ++t) {
    float xf = x0 + (float)(t & 1);
    float yf = y0 + (float)(t >> 1);
    float wv = ((t & 1) ? wx1 : wx0) * ((t >> 1) ? wy1 : wy0);
    bool valid = (xf >= 0.0f) && (xf <= (float)(W - 1)) && (yf >= 0.0f) && (yf <= (float)(H - 1));
    xs[t] = (int)fminf(fmaxf(xf, 0.0f), (float)(W - 1));
    ys[t] = (int)fminf(fmaxf(yf, 0.0f), (float)(H - 1));
    ws[t] = valid ? wv : 0.0f;
  }
}

// ---------------- kernel 0: reparameterized flow samples ----------------
__global__ __launch_bounds__(256) void sample_kernel(
    const float* __restrict__ meanf, const float* __restrict__ lvf, const float* __restrict__ nf,
    const float* __restrict__ meanb, const float* __restrict__ lvb, const float* __restrict__ nb,
    float* __restrict__ flowf, float* __restrict__ flowb) {
  long i = (long)blockIdx.x * 256 + threadIdx.x;   // one float4 per thread; grid covers exactly N/4
  const float4* m4f = (const float4*)meanf;  const float4* v4f = (const float4*)lvf;  const float4* n4f = (const float4*)nf;
  const float4* m4b = (const float4*)meanb;  const float4* v4b = (const float4*)lvb;  const float4* n4b = (const float4*)nb;
  float4 m = m4f[i], v = v4f[i], n = n4f[i];
  float4 o;
  o.x = m.x + __expf(0.5f * v.x) * n.x;  o.y = m.y + __expf(0.5f * v.y) * n.y;
  o.z = m.z + __expf(0.5f * v.z) * n.z;  o.w = m.w + __expf(0.5f * v.w) * n.w;
  ((float4*)flowf)[i] = o;
  m = m4b[i]; v = v4b[i]; n = n4b[i];
  o.x = m.x + __expf(0.5f * v.x) * n.x;  o.y = m.y + __expf(0.5f * v.y) * n.y;
  o.z = m.z + __expf(0.5f * v.z) * n.z;  o.w = m.w + __expf(0.5f * v.w) * n.w;
  ((float4*)flowb)[i] = o;
}

// ---------------- kernel 1: per-direction energy (tile + halo in LDS) ----------------
__global__ __launch_bounds__(256) void energy_kernel(
    const float* __restrict__ fa, const float* __restrict__ fb,
    const float* __restrict__ ia, const float* __restrict__ ib,
    float* __restrict__ partials) {
  __shared__ float s_ia[3][SH][SW];   // img_a tile: filled via async global->LDS DMA
  __shared__ float s_ib[3][SH][SW];   // warped img_b tile (computed, with halo)
  __shared__ float s_fa[2][SH][SW];
  __shared__ float red[256];

  const int tid = threadIdx.x;
  const int bx = blockIdx.x * TW, by = blockIdx.y * TH, b = blockIdx.z;

  // ---- stage: img_a via async-to-LDS, flow_a + bilinear-warped img_b via VALU ----
  for (int idx = tid; idx < SH * SW; idx += 256) {
    int sy = idx / SW, sx = idx - sy * SW;
    int gx = bx + sx - HL, gy = by + sy - HL;
    if (gx >= 0 && gx < W && gy >= 0 && gy < H) {
      long p = (long)gy * W + gx;
      // img_a plane rows -> LDS, asynchronously (overlaps with warp gather below)
      async_copy_b32(lds_off_of(&s_ia[0][sy][sx]), ia + (long)(b * 3 + 0) * HW + p);
      async_copy_b32(lds_off_of(&s_ia[1][sy][sx]), ia + (long)(b * 3 + 1) * HW + p);
      async_copy_b32(lds_off_of(&s_ia[2][sy][sx]), ia + (long)(b * 3 + 2) * HW + p);
      float f0 = fa[(long)(b * 2 + 0) * HW + p];
      float f1 = fa[(long)(b * 2 + 1) * HW + p];
      int xs[4], ys[4]; float wt[4];
      bilin_taps((float)gx + f0, (float)gy + f1, xs, ys, wt);
      float w0 = 0.f, w1 = 0.f, w2 = 0.f;
#pragma unroll
      for (int t = 0; t < 4; ++t) {
        long q = (long)ys[t] * W + xs[t];
        w0 += wt[t] * ib[(long)(b * 3 + 0) * HW + q];
        w1 += wt[t] * ib[(long)(b * 3 + 1) * HW + q];
        w2 += wt[t] * ib[(long)(b * 3 + 2) * HW + q];
      }
      s_fa[0][sy][sx] = f0;  s_fa[1][sy][sx] = f1;
      s_ib[0][sy][sx] = w0;  s_ib[1][sy][sx] = w1;  s_ib[2][sy][sx] = w2;
    } else {
      // zero padding for out-of-image halo (disjoint addresses from async writes)
      s_ia[0][sy][sx] = 0.f;  s_ia[1][sy][sx] = 0.f;  s_ia[2][sy][sx] = 0.f;
      s_fa[0][sy][sx] = 0.f;  s_fa[1][sy][sx] = 0.f;
      s_ib[0][sy][sx] = 0.f;  s_ib[1][sy][sx] = 0.f;  s_ib[2][sy][sx] = 0.f;
    }
  }
  async_wait0();        // ASYNCcnt == 0: all s_ia writes have landed in LDS
  __syncthreads();

  // ---- center pixels: all five energy terms ----
  float part = 0.0f;
  for (int p = tid; p < TW * TH; p += 256) {
    int cy = p / TW, cx = p - cy * TW;
    int sy = cy + HL, sx = cx + HL;
    int x = bx + cx, y = by + cy;

    float f0 = s_fa[0][sy][sx], f1 = s_fa[1][sy][sx];
    float xq = (float)x + f0, yq = (float)y + f1;

    // border mask
    float mx = sigf(xq + 0.5f) * (1.0f - sigf(xq - ((float)W - 0.5f)));
    float my = sigf(yq + 0.5f) * (1.0f - sigf(yq - ((float)H - 0.5f)));
    float bm = mx * my;

    // warp flow_b at (xq, yq)
    int xs[4], ys[4]; float wt[4];
    bilin_taps(xq, yq, xs, ys, wt);
    float fbw0 = 0.f, fbw1 = 0.f;
#pragma unroll
    for (int t = 0; t < 4; ++t) {
      long q = (long)ys[t] * W + xs[t];
      fbw0 += wt[t] * fb[(long)(b * 2 + 0) * HW + q];
      fbw1 += wt[t] * fb[(long)(b * 2 + 1) * HW + q];
    }

    float mag = f0 * f0 + f1 * f1 + fbw0 * fbw0 + fbw1 * fbw1;
    float d0 = f0 + fbw0, d1 = f1 + fbw1;
    float diff2 = d0 * d0 + d1 * d1;
    float occ = 1.0f - sigf(diff2 - (0.01f * mag + 0.5f));
    float m = bm * occ;

    part += (1.0f - m);                                  // mask term

    float e0 = s_ia[0][sy][sx] - s_ib[0][sy][sx];
    float e1 = s_ia[1][sy][sx] - s_ib[1][sy][sx];
    float e2 = s_ia[2][sy][sx] - s_ib[2][sy][sx];
    part += penf(e0 * e0 + e1 * e1 + e2 * e2) * m;       // data term
    part += penf(diff2) * m;                             // fwd/bwd consistency

    // smoothness (zero-padded at last row/col)
    float sdx = 0.0f, sdy = 0.0f;
    if (x < W - 1) {
      float u = s_fa[0][sy][sx + 1] - f0, v = s_fa[1][sy][sx + 1] - f1;
      sdx = u * u + v * v;
    }
    if (y < H - 1) {
      float u = s_fa[0][sy + 1][sx] - f0, v = s_fa[1][sy + 1][sx] - f1;
      sdy = u * u + v * v;
    }
    part += penf(sdx + sdy);                             // smoothness

    // gradient-constancy: 5-tap filters along W and H on img_a and warped img_b
    float Ct = 0.0f;
#pragma unroll
    for (int c = 0; c < 3; ++c) {
      float gwa = K0 * s_ia[c][sy][sx - 2] + K1 * s_ia[c][sy][sx - 1]
                + K3 * s_ia[c][sy][sx + 1] + K4 * s_ia[c][sy][sx + 2];
      float gwb = K0 * s_ib[c][sy][sx - 2] + K1 * s_ib[c][sy][sx - 1]
                + K3 * s_ib[c][sy][sx + 1] + K4 * s_ib[c][sy][sx + 2];
      float gha = K0 * s_ia[c][sy - 2][sx] + K1 * s_ia[c][sy - 1][sx]
                + K3 * s_ia[c][sy + 1][sx] + K4 * s_ia[c][sy + 2][sx];
      float ghb = K0 * s_ib[c][sy - 2][sx] + K1 * s_ib[c][sy - 1][sx]
                + K3 * s_ib[c][sy + 1][sx] + K4 * s_ib[c][sy + 2][sx];
      float dw = gwa - gwb, dh = gha - ghb;
      Ct += dw * dw + dh * dh;
    }
    part += penf(Ct) * m;                                // gradient term
  }

  // ---- block reduction ----
  red[tid] = part;
  __syncthreads();
  for (int s = 128; s > 0; s >>= 1) {
    if (tid < s) red[tid] += red[tid + s];
    __syncthreads();
  }
  if (tid == 0) {
    int blin = blockIdx.x + TILES_X * (blockIdx.y + TILES_Y * blockIdx.z);
    partials[blin] = red[0];
  }
}

// ---------------- kernel 2: EPE + entropy partials ----------------
__global__ __launch_bounds__(256) void misc_kernel(
    const float* __restrict__ meanf, const float* __restrict__ target,
    const float* __restrict__ lvf, const float* __restrict__ lvb,
    float* __restrict__ epeP, float* __restrict__ entP) {
  __shared__ float red[256];
  const int tid = threadIdx.x;
  const long N = (long)NB * HW;
  const long STRIDE = (long)NTILES * 256;
  float epe = 0.0f, ent = 0.0f;
  for (long i = (long)blockIdx.x * 256 + tid; i < N; i += STRIDE) {
    long b = i / HW, r = i - b * HW;
    long base = b * 2 * HW + r;
    if (i + STRIDE < N) {                    // prefetch next grid-stride chunk
      __builtin_prefetch(meanf + base + STRIDE, 0, 0);
      __builtin_prefetch(target + base + STRIDE, 0, 0);
    }
    float d0 = meanf[base] - target[base];
    float d1 = meanf[base + HW] - target[base + HW];
    epe += sqrtf(d0 * d0 + d1 * d1);
    ent += lvf[base] + lvf[base + HW] + lvb[base] + lvb[base + HW];
  }
  red[tid] = epe;
  __syncthreads();
  for (int s = 128; s > 0; s >>= 1) { if (tid < s) red[tid] += red[tid + s]; __syncthreads(); }
  if (tid == 0) epeP[blockIdx.x] = red[0];
  __syncthreads();
  red[tid] = ent;
  __syncthreads();
  for (int s = 128; s > 0; s >>= 1) { if (tid < s) red[tid] += red[tid + s]; __syncthreads(); }
  if (tid == 0) entP[blockIdx.x] = red[0];
}

// ---------------- kernel 3: WMMA-based final reduction + scalar combine ----------------
// D = A x B + C with A = 16x4 all-ones: D[m][n] = sum_k B[k][n], so every WMMA
// folds 64 fp32 partials into the f32 accumulator; sum(all D)/16 == total.
__device__ __forceinline__ float wmma_reduce(const float* __restrict__ p, int n, int lane) {
  v2f a; a[0] = 1.0f; a[1] = 1.0f;
  v8f acc = {};
  for (int i = 0; i < n; i += 64) {
    v2f bv;
    bv[0] = p[i + 2 * lane];
    bv[1] = p[i + 2 * lane + 1];
    acc = __builtin_amdgcn_wmma_f32_16x16x4_f32(false, a, false, bv, (short)0, acc, false, false);
  }
  float s = acc[0] + acc[1] + acc[2] + acc[3] + acc[4] + acc[5] + acc[6] + acc[7];
#pragma unroll
  for (int off = 16; off > 0; off >>= 1) s += __shfl_down(s, off, 32);
  s = __shfl(s, 0, 32);
  return s * (1.0f / 16.0f);
}

__global__ __launch_bounds__(32) void finalize_kernel(
    const float* __restrict__ eP, const float* __restrict__ epeP, const float* __restrict__ entP,
    float* __restrict__ out) {
  int lane = threadIdx.x;                 // one full wave32, EXEC all ones
  float eSum   = wmma_reduce(eP,   NPART_E, lane);
  float epeSum = wmma_reduce(epeP, NPART_M, lane);
  float entSum = wmma_reduce(entP, NPART_M, lane);
  if (lane == 0) {
    float elbo = (eSum - 0.5f * entSum) / (float)NB;
    float epe  = epeSum / ((float)NB * (float)HW);
    out[0] = elbo;
    out[1] = epe;
  }
}

// ---------------- host-side launcher ----------------
extern "C" void kernel_launch(void* const* d_in, const int* in_sizes, int n_in,
                              void* d_out, int out_size, void* d_ws, size_t ws_size,
                              hipStream_t stream) {
  const float* meanf  = (const float*)d_in[0];
  const float* lvf    = (const float*)d_in[1];
  const float* meanb  = (const float*)d_in[2];
  const float* lvb    = (const float*)d_in[3];
  const float* img1   = (const float*)d_in[4];
  const float* img2   = (const float*)d_in[5];
  const float* target = (const float*)d_in[6];
  const float* nf     = (const float*)d_in[7];
  const float* nb     = (const float*)d_in[8];

  float* ws = (float*)d_ws;
  const long NFLOW = (long)NB * 2 * HW;        // 3,145,728 floats per flow field
  float* flowf = ws;
  float* flowb = ws + NFLOW;
  float* eP    = ws + 2 * NFLOW;               // 6144
  float* epeP  = eP + NPART_E;                 // 3072
  float* entP  = epeP + NPART_M;               // 3072

  // 0) sample flows (exactly N/4 float4 per thread, 3072 blocks)
  sample_kernel<<<(int)(NFLOW / 4 / 256), 256, 0, stream>>>(
      meanf, lvf, nf, meanb, lvb, nb, flowf, flowb);

  // 1) energy, both directions
  dim3 egrid(TILES_X, TILES_Y, NB);
  energy_kernel<<<egrid, 256, 0, stream>>>(flowf, flowb, img1, img2, eP);            // fwd
  energy_kernel<<<egrid, 256, 0, stream>>>(flowb, flowf, img2, img1, eP + NTILES);   // bwd

  // 2) epe + entropy partials
  misc_kernel<<<NTILES, 256, 0, stream>>>(meanf, target, lvf, lvb, epeP, entP);

  // 3) WMMA reduction + combine
  finalize_kernel<<<1, 32, 0, stream>>>(eP, epeP, entP, (float*)d_out);
}